// LowRankFNO3d_51376398795194
// MI455X (gfx1250) — compile-verified
//
#include <hip/hip_runtime.h>
#include <math.h>

typedef __attribute__((ext_vector_type(2))) float v2f;
typedef __attribute__((ext_vector_type(8))) float v8f;

#define BATCH 4
#define NX 70
#define NY 70
#define NZ 46
#define CH 32
#define RK 8
#define MD 8
#define IX 64
#define IY 64
#define IZ 40
#define FC1N 128

#define H_SZ   (BATCH*NX*NY*NZ*CH)      /* 28,851,200 floats */
#define HZ_SZ  (BATCH*NX*NY*MD*CH*2)    /* 10,035,200 floats (aliased with G) */
#define HYZ_SZ (BATCH*NX*MD*MD*CH*2)    /* 1,146,880 floats  (aliased with Gx) */
#define FM_SZ  (BATCH*MD*MD*MD*CH*2)    /* 131,072 floats */
#define YM_SZ  FM_SZ

#define TWO_PI 6.28318530717958647692f

__device__ inline v8f wmma4(v2f a, v2f b, v8f c) {
  // D = A(16x4 f32) * B(4x16 f32) + C(16x16 f32)
  return __builtin_amdgcn_wmma_f32_16x16x4_f32(false, a, false, b, (short)0, c,
                                               false, false);
}

__device__ inline float gelu_f(float v) {
  return 0.5f * v * (1.0f + erff(v * 0.70710678118654752440f));
}

// ---------------------------------------------------------------------------
// fc0: x(b,64,64,40,4) @ W(4,32) + b -> interior of padded h(b,70,70,46,32)
// One wave = 16 points, K=4 => a single WMMA k-step per 16x16 output tile.
// ---------------------------------------------------------------------------
__global__ void fc0_kernel(const float* __restrict__ x, const float* __restrict__ w,
                           const float* __restrict__ bvec, float* __restrict__ h) {
  int lane = threadIdx.x & 31;
  int wid  = (blockIdx.x * blockDim.x + threadIdx.x) >> 5;
  int m = lane & 15, hi = lane >> 4;
  int kb = hi * 2;
  int p0 = wid * 16;
  v2f a;
  { int p = p0 + m; a.x = x[p * 4 + kb]; a.y = x[p * 4 + kb + 1]; }
#pragma unroll
  for (int nt = 0; nt < 2; ++nt) {
    int n = nt * 16 + m;
    v2f bf; bf.x = w[kb * CH + n]; bf.y = w[(kb + 1) * CH + n];
    v8f c = {};
    c = wmma4(a, bf, c);
    float bias = bvec[n];
#pragma unroll
    for (int j = 0; j < 8; ++j) {
      int pp = p0 + j + hi * 8;
      int zi = pp % IZ; int t = pp / IZ;
      int yi = t % IY;  t /= IY;
      int xi = t % IX;  int bb = t / IX;
      h[(((bb * NX + xi) * NY + yi) * NZ + zi) * CH + n] = c[j] + bias;
    }
  }
}

// ---------------------------------------------------------------------------
// Forward partial DFT, stage Z: h(b,x,y,z,c) -> Hz(b,x,y,k3<8,c) complex
// thread = (b,x,y,c); 8 complex accumulators; twiddle power recurrence.
// ---------------------------------------------------------------------------
__global__ void dft_z_kernel(const float* __restrict__ h, float* __restrict__ Hz) {
  int t = blockIdx.x * blockDim.x + threadIdx.x;
  int c = t & 31; int q = t >> 5;                   // q = (b*NX+x)*NY+y
  const float* row = h + (size_t)q * NZ * CH + c;
  float ar[MD] = {}, ai[MD] = {};
  for (int z = 0; z < NZ; ++z) {
    float v = row[z * CH];
    float s, co; __sincosf(-TWO_PI * (float)z / (float)NZ, &s, &co);
    float wr = 1.f, wi = 0.f;
#pragma unroll
    for (int k = 0; k < MD; ++k) {
      ar[k] += v * wr; ai[k] += v * wi;
      float nr = wr * co - wi * s; wi = wr * s + wi * co; wr = nr;
    }
  }
  float* out = Hz + (size_t)q * MD * CH * 2;
#pragma unroll
  for (int k = 0; k < MD; ++k) {
    out[(k * CH + c) * 2]     = ar[k];
    out[(k * CH + c) * 2 + 1] = ai[k];
  }
}

// Stage Y: Hz(b,x,y,k3,c) -> Hyz(b,x,k2<8,k3,c); thread=(b,x,k3,c), loop y.
__global__ void dft_y_kernel(const float* __restrict__ Hz, float* __restrict__ Hyz) {
  int t = blockIdx.x * blockDim.x + threadIdx.x;
  int c = t & 31; int q = t >> 5;
  int k3 = q & 7; int xb = q >> 3;                  // xb = b*NX + x
  const float* in = Hz + ((((size_t)xb * NY) * MD + k3) * CH + c) * 2;
  float ar[MD] = {}, ai[MD] = {};
  for (int y = 0; y < NY; ++y) {
    float vr = in[(size_t)y * MD * CH * 2];
    float vi = in[(size_t)y * MD * CH * 2 + 1];
    float s, co; __sincosf(-TWO_PI * (float)y / (float)NY, &s, &co);
    float wr = 1.f, wi = 0.f;
#pragma unroll
    for (int k = 0; k < MD; ++k) {
      ar[k] += vr * wr - vi * wi; ai[k] += vr * wi + vi * wr;
      float nr = wr * co - wi * s; wi = wr * s + wi * co; wr = nr;
    }
  }
#pragma unroll
  for (int k2 = 0; k2 < MD; ++k2) {
    float* o = Hyz + ((((size_t)xb * MD + k2) * MD + k3) * CH + c) * 2;
    o[0] = ar[k2]; o[1] = ai[k2];
  }
}

// Stage X: Hyz(b,x,k2,k3,c) -> Fm(b,k1<8,k2,k3,c); thread=(b,k2,k3,c), loop x.
__global__ void dft_x_kernel(const float* __restrict__ Hyz, float* __restrict__ Fm) {
  int t = blockIdx.x * blockDim.x + threadIdx.x;
  int c = t & 31; int q = t >> 5;
  int k3 = q & 7; int k2 = (q >> 3) & 7; int b = q >> 6;
  const float* in = Hyz + ((((size_t)b * NX * MD + k2) * MD + k3) * CH + c) * 2;
  float ar[MD] = {}, ai[MD] = {};
  for (int x = 0; x < NX; ++x) {
    float vr = in[(size_t)x * MD * MD * CH * 2];
    float vi = in[(size_t)x * MD * MD * CH * 2 + 1];
    float s, co; __sincosf(-TWO_PI * (float)x / (float)NX, &s, &co);
    float wr = 1.f, wi = 0.f;
#pragma unroll
    for (int k = 0; k < MD; ++k) {
      ar[k] += vr * wr - vi * wi; ai[k] += vr * wi + vi * wr;
      float nr = wr * co - wi * s; wi = wr * s + wi * co; wr = nr;
    }
  }
#pragma unroll
  for (int k1 = 0; k1 < MD; ++k1) {
    float* o = Fm + (((((size_t)b * MD + k1) * MD + k2) * MD + k3) * CH + c) * 2;
    o[0] = ar[k1]; o[1] = ai[k1];
  }
}

// ---------------------------------------------------------------------------
// Low-rank mode mixing: xr = U^T F (32->8), * Bc, back with V (8->32).
// Tiny (2048 modes); one thread per mode.
// ---------------------------------------------------------------------------
__global__ void mix_kernel(const float* __restrict__ Fm, const float* __restrict__ U,
                           const float* __restrict__ V, const float* __restrict__ Br,
                           const float* __restrict__ Bi, float* __restrict__ Ym) {
  int t = blockIdx.x * blockDim.x + threadIdx.x;     // t = b*512+k1*64+k2*8+k3
  int moff = t & 511;                                // k1*64+k2*8+k3
  const float* f = Fm + (size_t)t * CH * 2;
  float xr[RK] = {}, xi[RK] = {};
  for (int c = 0; c < CH; ++c) {
    float fr = f[c * 2], fi = f[c * 2 + 1];
#pragma unroll
    for (int r = 0; r < RK; ++r) {
      float u = U[c * RK + r];
      xr[r] += fr * u; xi[r] += fi * u;
    }
  }
#pragma unroll
  for (int r = 0; r < RK; ++r) {
    float br = Br[r * 512 + moff], bi = Bi[r * 512 + moff];
    float tr = xr[r] * br - xi[r] * bi;
    float ti = xr[r] * bi + xi[r] * br;
    xr[r] = tr; xi[r] = ti;
  }
  float* yo = Ym + (size_t)t * CH * 2;
  for (int o = 0; o < CH; ++o) {
    float ar = 0.f, ai = 0.f;
#pragma unroll
    for (int r = 0; r < RK; ++r) {
      float v = V[o * RK + r];
      ar += xr[r] * v; ai += xi[r] * v;
    }
    yo[o * 2] = ar; yo[o * 2 + 1] = ai;
  }
}

// Inverse stage X: Ym(b,k1,k2,k3,c) -> Gx(b,x,k2,k3,c); thread=(b,x,k2,k3,c).
__global__ void idft_x_kernel(const float* __restrict__ Ym, float* __restrict__ Gx) {
  int t = blockIdx.x * blockDim.x + threadIdx.x;
  int c = t & 31; int q = t >> 5;
  int k3 = q & 7; int k2 = (q >> 3) & 7; int x = (q >> 6) % NX; int b = (q >> 6) / NX;
  const float* in = Ym + ((size_t)(b * 512 + k2 * 8 + k3) * CH + c) * 2;
  float s, co; __sincosf(TWO_PI * (float)x / (float)NX, &s, &co);
  float wr = 1.f, wi = 0.f, ar = 0.f, ai = 0.f;
#pragma unroll
  for (int k1 = 0; k1 < MD; ++k1) {
    float vr = in[(size_t)k1 * MD * MD * CH * 2];
    float vi = in[(size_t)k1 * MD * MD * CH * 2 + 1];
    ar += vr * wr - vi * wi; ai += vr * wi + vi * wr;
    float nr = wr * co - wi * s; wi = wr * s + wi * co; wr = nr;
  }
  float* o = Gx + ((((size_t)(b * NX + x) * MD + k2) * MD + k3) * CH + c) * 2;
  o[0] = ar; o[1] = ai;
}

// Inverse stage Y: Gx(b,x,k2,k3,c) -> G(b,x,y,k3,c); thread=(b,x,y,k3,c).
__global__ void idft_y_kernel(const float* __restrict__ Gx, float* __restrict__ G) {
  int t = blockIdx.x * blockDim.x + threadIdx.x;
  int c = t & 31; int q = t >> 5;
  int k3 = q & 7; int r2 = q >> 3;
  int y = r2 % NY; int r3 = r2 / NY;
  int x = r3 % NX; int b = r3 / NX;
  const float* in = Gx + (((size_t)(b * NX + x) * MD * MD + k3) * CH + c) * 2;
  float s, co; __sincosf(TWO_PI * (float)y / (float)NY, &s, &co);
  float wr = 1.f, wi = 0.f, ar = 0.f, ai = 0.f;
#pragma unroll
  for (int k2 = 0; k2 < MD; ++k2) {
    float vr = in[(size_t)k2 * MD * CH * 2];
    float vi = in[(size_t)k2 * MD * CH * 2 + 1];
    ar += vr * wr - vi * wi; ai += vr * wi + vi * wr;
    float nr = wr * co - wi * s; wi = wr * s + wi * co; wr = nr;
  }
  float* o = G + ((((size_t)(b * NX + x) * NY + y) * MD + k3) * CH + c) * 2;
  o[0] = ar; o[1] = ai;
}

// ---------------------------------------------------------------------------
// Fused layer epilogue: inverse-z DFT (as a 16x16xK=16 WMMA GEMM with
// normalization + Hermitian weights folded into the twiddle A-matrix) plus
// the 32x32 pointwise conv (K=32 WMMA, A staged in LDS) accumulated into the
// SAME WMMA accumulators, then bias + GeLU, in-place update of h.
// One wave per (b,x,y) row.
// ---------------------------------------------------------------------------
__global__ void layer_fuse_kernel(const float* __restrict__ G,
                                  const float* __restrict__ Wm,
                                  const float* __restrict__ Wbv,
                                  float* __restrict__ h, int apply_gelu) {
  __shared__ float sh_all[8 * NZ * CH];              // 47,104 B static LDS
  int lane = threadIdx.x & 31;
  int wib = threadIdx.x >> 5;
  int wid = blockIdx.x * 8 + wib;                    // (b*NX+x)*NY+y
  float* hrow = h + (size_t)wid * NZ * CH;
  const float* grow = G + (size_t)wid * MD * CH * 2;
  float* sh = sh_all + wib * NZ * CH;
  for (int z = 0; z < NZ; ++z) sh[z * CH + lane] = hrow[z * CH + lane];
  __syncthreads();

  int m = lane & 15, hi = lane >> 4;
  const float invS = 1.0f / (float)(NX * NY * NZ);

  for (int zt = 0; zt < 3; ++zt) {
    v8f acc0 = {}; v8f acc1 = {};
    int z = zt * 16 + m;
    bool zok = (z < NZ);

    // ---- inverse-z DFT: K = 16 (rows 0..7 = Re modes, 8..15 = Im modes)
#pragma unroll
    for (int kk = 0; kk < 4; ++kk) {
      int kr0 = kk * 4 + hi * 2;
      v2f a;
#pragma unroll
      for (int u = 0; u < 2; ++u) {
        int k = kr0 + u;
        int k3 = k & 7;
        float wk = (k3 == 0) ? 1.f : 2.f;
        float s, co;
        __sincosf(TWO_PI * (float)(k3 * z) / (float)NZ, &s, &co);
        float val = (k < 8) ? invS * wk * co : -invS * wk * s;
        if (!zok) val = 0.f;
        if (u == 0) a.x = val; else a.y = val;
      }
      v2f b0, b1;
      b0.x = (kr0     < 8) ? grow[(kr0 * CH + m) * 2]            : grow[((kr0 - 8) * CH + m) * 2 + 1];
      b0.y = (kr0 + 1 < 8) ? grow[((kr0 + 1) * CH + m) * 2]      : grow[((kr0 - 7) * CH + m) * 2 + 1];
      b1.x = (kr0     < 8) ? grow[(kr0 * CH + 16 + m) * 2]       : grow[((kr0 - 8) * CH + 16 + m) * 2 + 1];
      b1.y = (kr0 + 1 < 8) ? grow[((kr0 + 1) * CH + 16 + m) * 2] : grow[((kr0 - 7) * CH + 16 + m) * 2 + 1];
      acc0 = wmma4(a, b0, acc0);
      acc1 = wmma4(a, b1, acc1);
    }

    // ---- pointwise 32x32 conv: K = 32, A rows from LDS copy of h
#pragma unroll
    for (int kk = 0; kk < 8; ++kk) {
      int kr0 = kk * 4 + hi * 2;
      v2f a;
      a.x = zok ? sh[z * CH + kr0]     : 0.f;
      a.y = zok ? sh[z * CH + kr0 + 1] : 0.f;
      v2f b0, b1;                                    // B[k][n] = Ww[n][k]
      b0.x = Wm[m * CH + kr0];        b0.y = Wm[m * CH + kr0 + 1];
      b1.x = Wm[(16 + m) * CH + kr0]; b1.y = Wm[(16 + m) * CH + kr0 + 1];
      acc0 = wmma4(a, b0, acc0);
      acc1 = wmma4(a, b1, acc1);
    }

    float bias0 = Wbv[m], bias1 = Wbv[16 + m];
#pragma unroll
    for (int j = 0; j < 8; ++j) {
      int zz = zt * 16 + j + hi * 8;
      if (zz < NZ) {
        float v0 = acc0[j] + bias0;
        float v1 = acc1[j] + bias1;
        if (apply_gelu) { v0 = gelu_f(v0); v1 = gelu_f(v1); }
        hrow[zz * CH + m]      = v0;
        hrow[zz * CH + 16 + m] = v1;
      }
    }
  }
}

// ---------------------------------------------------------------------------
// Head: fc1 (32->128, WMMA) + GeLU + fc2 (128->1, lane dot + shfl butterfly).
// One wave = 16 interior points.
// ---------------------------------------------------------------------------
__global__ void head_kernel(const float* __restrict__ h, const float* __restrict__ w1,
                            const float* __restrict__ b1, const float* __restrict__ w2,
                            const float* __restrict__ b2, float* __restrict__ out) {
  int lane = threadIdx.x & 31;
  int wid  = (blockIdx.x * blockDim.x + threadIdx.x) >> 5;
  int m = lane & 15, hi = lane >> 4;
  int p0 = wid * 16;
  int pa = p0 + m;
  int zi = pa % IZ; int t = pa / IZ;
  int yi = t % IY;  t /= IY;
  int xi = t % IX;  int bb = t / IX;
  const float* arow = h + (size_t)(((bb * NX + xi) * NY + yi) * NZ + zi) * CH;

  float outacc[8] = {};
#pragma unroll
  for (int nt = 0; nt < 8; ++nt) {
    v8f acc = {};
    int n = nt * 16 + m;
#pragma unroll
    for (int kk = 0; kk < 8; ++kk) {
      int kr0 = kk * 4 + hi * 2;
      v2f a;  a.x = arow[kr0];              a.y = arow[kr0 + 1];
      v2f bf; bf.x = w1[kr0 * FC1N + n];    bf.y = w1[(kr0 + 1) * FC1N + n];
      acc = wmma4(a, bf, acc);
    }
    float bias = b1[n], w2n = w2[n];
#pragma unroll
    for (int j = 0; j < 8; ++j) {
      float v = gelu_f(acc[j] + bias);
      outacc[j] += v * w2n;
    }
  }
  float c2 = b2[0];
#pragma unroll
  for (int j = 0; j < 8; ++j) {
    float v = outacc[j];
    v += __shfl_xor(v, 8, 32);
    v += __shfl_xor(v, 4, 32);
    v += __shfl_xor(v, 2, 32);
    v += __shfl_xor(v, 1, 32);
    if (m == 0) out[p0 + j + hi * 8] = v + c2;
  }
}

// ---------------------------------------------------------------------------
extern "C" void kernel_launch(void* const* d_in, const int* in_sizes, int n_in,
                              void* d_out, int out_size, void* d_ws, size_t ws_size,
                              hipStream_t stream) {
  const float* x     = (const float*)d_in[0];
  const float* fc0_w = (const float*)d_in[1];
  const float* fc0_b = (const float*)d_in[2];
  const float* Us    = (const float*)d_in[3];
  const float* Vs    = (const float*)d_in[4];
  const float* Bre   = (const float*)d_in[5];
  const float* Bim   = (const float*)d_in[6];
  const float* Ww    = (const float*)d_in[7];
  const float* Wb    = (const float*)d_in[8];
  const float* fc1_w = (const float*)d_in[9];
  const float* fc1_b = (const float*)d_in[10];
  const float* fc2_w = (const float*)d_in[11];
  const float* fc2_b = (const float*)d_in[12];
  (void)in_sizes; (void)n_in; (void)out_size; (void)ws_size;

  float* ws  = (float*)d_ws;
  float* h   = ws;                 // 28,851,200 floats
  float* Hz  = h + H_SZ;           // 10,035,200 floats (also G)
  float* Hyz = Hz + HZ_SZ;         // 1,146,880 floats  (also Gx)
  float* Fm  = Hyz + HYZ_SZ;       // 131,072 floats
  float* Ym  = Fm + FM_SZ;         // 131,072 floats

  hipMemsetAsync(h, 0, (size_t)H_SZ * sizeof(float), stream);   // zero pad region
  fc0_kernel<<<5120, 256, 0, stream>>>(x, fc0_w, fc0_b, h);     // 40,960 waves

  for (int i = 0; i < 4; ++i) {
    dft_z_kernel <<<2450, 256, 0, stream>>>(h, Hz);
    dft_y_kernel <<<280,  256, 0, stream>>>(Hz, Hyz);
    dft_x_kernel <<<32,   256, 0, stream>>>(Hyz, Fm);
    mix_kernel   <<<8,    256, 0, stream>>>(Fm, Us + i * CH * RK, Vs + i * CH * RK,
                                            Bre + i * RK * 512, Bim + i * RK * 512, Ym);
    idft_x_kernel<<<2240, 256, 0, stream>>>(Ym, Hyz);           // Gx aliases Hyz
    idft_y_kernel<<<19600,256, 0, stream>>>(Hyz, Hz);           // G  aliases Hz
    layer_fuse_kernel<<<2450, 256, 0, stream>>>(Hz, Ww + i * CH * CH, Wb + i * CH,
                                                h, (i < 3) ? 1 : 0);
  }

  head_kernel<<<5120, 256, 0, stream>>>(h, fc1_w, fc1_b, fc2_w, fc2_b, (float*)d_out);
}